// Actor_77068893159548
// MI455X (gfx1250) — compile-verified
//
#include <hip/hip_runtime.h>

#define N_NODES 50000
#define N_EDGES 800000
#define N_GRAPHS 256
#define IN_DIM 64
#define OUT_DIM 128
#define ETOT (N_EDGES + N_NODES)
#define NEG_SLOPE 0.2f

// workspace layout (float offsets)
#define WS_X    0                                   // [N,128] transformed features
#define WS_ASRC (N_NODES * 128)                     // [N,8]
#define WS_ADST (WS_ASRC + N_NODES * 8)             // [N,8]
#define WS_M    (WS_ADST + N_NODES * 8)             // [N,8] segment max
#define WS_S    (WS_M + N_NODES * 8)                // [N,8] segment expsum
#define WS_ACC  (WS_S + N_NODES * 8)                // [N,128] aggregate -> op_embedding (in place)
#define WS_GSUM (WS_ACC + N_NODES * 128)            // [G,128]
#define WS_GW   (WS_GSUM + N_GRAPHS * 128)          // [G,128] graph_emb@W1_bot + b1
#define WS_GCNT (WS_GW + N_GRAPHS * 128)            // [G]

typedef __attribute__((ext_vector_type(2))) float v2f;
typedef __attribute__((ext_vector_type(8))) float v8f;

__device__ __forceinline__ v8f wmma4(v2f a, v2f b, v8f c) {
  // D = A(16x4 f32) x B(4x16 f32) + C(16x16 f32)
  return __builtin_amdgcn_wmma_f32_16x16x4_f32(false, a, false, b, (short)0, c,
                                               false, false);
}

__device__ __forceinline__ float lane_reduce16(float v) {
  v += __shfl_xor(v, 1, 32);
  v += __shfl_xor(v, 2, 32);
  v += __shfl_xor(v, 4, 32);
  v += __shfl_xor(v, 8, 32);
  return v;
}

__device__ __forceinline__ void atomicMaxF(float* addr, float val) {
  // standard sign-split trick; works with -inf init
  if (val >= 0.0f) atomicMax((int*)addr, __float_as_int(val));
  else             atomicMin((unsigned int*)addr, __float_as_uint(val));
}

// ---------------- init ----------------
__global__ void k_init(float* __restrict__ ws) {
  int i = blockIdx.x * blockDim.x + threadIdx.x;
  if (i < N_NODES * 128) ws[WS_ACC + i] = 0.0f;
  if (i < N_NODES * 8) { ws[WS_M + i] = -__builtin_inff(); ws[WS_S + i] = 0.0f; }
  if (i < N_GRAPHS * 128) ws[WS_GSUM + i] = 0.0f;
  if (i < N_GRAPHS) ws[WS_GCNT + i] = 0.0f;
}

// ---------------- GAT linear: x = feat @ W_gat, + a_src/a_dst ----------------
// grid = 3125 blocks x 256 thr (8 waves). wave w = head h = col tile; block = 16 rows.
__global__ void k_gat_gemm(const float* __restrict__ feat, const float* __restrict__ Wg,
                           const float* __restrict__ att_s, const float* __restrict__ att_d,
                           float* __restrict__ ws) {
  const int h    = threadIdx.x >> 5;        // head / col tile 0..7
  const int lane = threadIdx.x & 31;
  const int half = lane >> 4, lp = lane & 15;
  const int m0   = blockIdx.x * 16;
  const int col  = h * 16 + lp;
  const int arow = m0 + lp;
  v8f acc = {0.f, 0.f, 0.f, 0.f, 0.f, 0.f, 0.f, 0.f};
#pragma unroll
  for (int k0 = 0; k0 < IN_DIM; k0 += 4) {
    const int kb = k0 + 2 * half;
    v2f a = *(const v2f*)(feat + arow * IN_DIM + kb);  // 8B-aligned (kb even)
    v2f b;
    b.x = Wg[kb * 128 + col];
    b.y = Wg[(kb + 1) * 128 + col];
    acc = wmma4(a, b, acc);
  }
  const float asv = att_s[h * 16 + lp];
  const float adv = att_d[h * 16 + lp];
  float* X = ws + WS_X;
#pragma unroll
  for (int i = 0; i < 8; ++i) {
    const int r = m0 + i + 8 * half;
    X[r * 128 + col] = acc[i];
    float ps = lane_reduce16(acc[i] * asv);
    float pd = lane_reduce16(acc[i] * adv);
    if (lp == 0) {
      ws[WS_ASRC + r * 8 + h] = ps;
      ws[WS_ADST + r * 8 + h] = pd;
    }
  }
}

// ---------------- edge pass 1: segment max ----------------
__global__ void k_edge_max(const int* __restrict__ ei, float* __restrict__ ws) {
  int idx = blockIdx.x * blockDim.x + threadIdx.x;  // ETOT*8 = 6.8M
  if (idx >= ETOT * 8) return;
  int e = idx >> 3, h = idx & 7;
  int src, dst;
  if (e < N_EDGES) { src = ei[e]; dst = ei[N_EDGES + e]; }
  else             { src = dst = e - N_EDGES; }
  float a = ws[WS_ASRC + src * 8 + h] + ws[WS_ADST + dst * 8 + h];
  a = a > 0.0f ? a : a * NEG_SLOPE;
  atomicMaxF(ws + WS_M + dst * 8 + h, a);
}

// ---------------- edge pass 2: exp-sum ----------------
__global__ void k_edge_sum(const int* __restrict__ ei, float* __restrict__ ws) {
  int idx = blockIdx.x * blockDim.x + threadIdx.x;
  if (idx >= ETOT * 8) return;
  int e = idx >> 3, h = idx & 7;
  int src, dst;
  if (e < N_EDGES) { src = ei[e]; dst = ei[N_EDGES + e]; }
  else             { src = dst = e - N_EDGES; }
  float a = ws[WS_ASRC + src * 8 + h] + ws[WS_ADST + dst * 8 + h];
  a = a > 0.0f ? a : a * NEG_SLOPE;
  float ex = __expf(a - ws[WS_M + dst * 8 + h]);
  atomicAdd(ws + WS_S + dst * 8 + h, ex);
}

// ---------------- edge pass 3: weighted aggregate (wave per edge) ----------------
__global__ void k_edge_aggr(const int* __restrict__ ei, float* __restrict__ ws) {
  int wid = (blockIdx.x * blockDim.x + threadIdx.x) >> 5;
  if (wid >= ETOT) return;
  int lane = threadIdx.x & 31;
  int src, dst;
  if (wid < N_EDGES) { src = ei[wid]; dst = ei[N_EDGES + wid]; }
  else               { src = dst = wid - N_EDGES; }
  const int h  = lane >> 2;       // 4 lanes per head
  const int ch = lane * 4;        // 4 channels per lane
  float a = ws[WS_ASRC + src * 8 + h] + ws[WS_ADST + dst * 8 + h];
  a = a > 0.0f ? a : a * NEG_SLOPE;
  float w = __expf(a - ws[WS_M + dst * 8 + h]) / ws[WS_S + dst * 8 + h];
  float4 xs = *(const float4*)(ws + WS_X + src * 128 + ch);
  float* o = ws + WS_ACC + dst * 128 + ch;
  atomicAdd(o + 0, w * xs.x);
  atomicAdd(o + 1, w * xs.y);
  atomicAdd(o + 2, w * xs.z);
  atomicAdd(o + 3, w * xs.w);
}

// ---------------- ELU + pooled sums (batch sorted -> register accumulation) ----------------
__global__ void k_finalize(const int* __restrict__ batch, const float* __restrict__ bias,
                           float* __restrict__ ws) {
  const int c  = threadIdx.x;        // 0..127
  const int n0 = blockIdx.x * 32;
  float sum = 0.0f;
  int gcur = batch[n0];
  for (int j = 0; j < 32; ++j) {
    int n = n0 + j;
    if (n >= N_NODES) break;
    float v = ws[WS_ACC + n * 128 + c] + bias[c];
    v = v > 0.0f ? v : __expf(v) - 1.0f;     // ELU
    ws[WS_ACC + n * 128 + c] = v;            // becomes op_embedding in place
    int g = batch[n];
    if (g != gcur) { atomicAdd(ws + WS_GSUM + gcur * 128 + c, sum); sum = 0.0f; gcur = g; }
    sum += v;
  }
  atomicAdd(ws + WS_GSUM + gcur * 128 + c, sum);
}

__global__ void k_count(const int* __restrict__ batch, float* __restrict__ ws) {
  int n = blockIdx.x * blockDim.x + threadIdx.x;
  if (n < N_NODES) atomicAdd(ws + WS_GCNT + batch[n], 1.0f);
}

// ---------------- mean divide -> graph_embedding output ----------------
__global__ void k_pool_div(const float* __restrict__ ws, float* __restrict__ out) {
  int i = blockIdx.x * blockDim.x + threadIdx.x;
  if (i >= N_GRAPHS * 128) return;
  float c = ws[WS_GCNT + (i >> 7)];
  out[N_NODES + i] = ws[WS_GSUM + i] / fmaxf(c, 1.0f);
}

// ---------------- gW = graph_emb @ W1[128:256,:] + b1  (WMMA) ----------------
// grid 16 x 256 (8 waves = col tiles)
__global__ void k_gw(const float* __restrict__ gemb, const float* __restrict__ W1,
                     const float* __restrict__ b1, float* __restrict__ ws) {
  const int wv   = threadIdx.x >> 5;
  const int lane = threadIdx.x & 31;
  const int half = lane >> 4, lp = lane & 15;
  const int m0   = blockIdx.x * 16;
  const int col  = wv * 16 + lp;
  const int arow = m0 + lp;
  v8f acc = {0.f, 0.f, 0.f, 0.f, 0.f, 0.f, 0.f, 0.f};
#pragma unroll
  for (int k0 = 0; k0 < 128; k0 += 4) {
    const int kb = k0 + 2 * half;
    v2f a = *(const v2f*)(gemb + arow * 128 + kb);
    v2f b;
    b.x = W1[(128 + kb) * 128 + col];
    b.y = W1[(129 + kb) * 128 + col];
    acc = wmma4(a, b, acc);
  }
  const float bv = b1[col];
#pragma unroll
  for (int i = 0; i < 8; ++i) {
    const int r = m0 + i + 8 * half;
    ws[WS_GW + r * 128 + col] = acc[i] + bv;
  }
}

// ---------------- decoder: h = relu(op_emb@W1_top + gW[batch]); scores = h@W2 + b2 ----------------
// grid 625 x 160 (5 waves/block, 3125 row tiles exactly -> EXEC all ones at WMMA)
__global__ void k_decode(const float* __restrict__ W1, const float* __restrict__ W2,
                         const float* __restrict__ b2, const int* __restrict__ batch,
                         const float* __restrict__ ws, float* __restrict__ out) {
  const int t    = blockIdx.x * 5 + (threadIdx.x >> 5);  // row tile 0..3124
  const int lane = threadIdx.x & 31;
  const int half = lane >> 4, lp = lane & 15;
  const int m0   = t * 16;
  const float* emb = ws + WS_ACC;
  v8f acc[8];
#pragma unroll
  for (int j = 0; j < 8; ++j) acc[j] = (v8f){0.f, 0.f, 0.f, 0.f, 0.f, 0.f, 0.f, 0.f};
  const int arow = m0 + lp;
  for (int k0 = 0; k0 < 128; k0 += 4) {
    const int kb = k0 + 2 * half;
    v2f a = *(const v2f*)(emb + arow * 128 + kb);
#pragma unroll
    for (int j = 0; j < 8; ++j) {
      v2f b;
      b.x = W1[kb * 128 + j * 16 + lp];
      b.y = W1[(kb + 1) * 128 + j * 16 + lp];
      acc[j] = wmma4(a, b, acc[j]);
    }
  }
  const float b2v = b2[0];
#pragma unroll
  for (int i = 0; i < 8; ++i) {
    const int r = m0 + i + 8 * half;
    const int g = batch[r];
    float sc = 0.0f;
#pragma unroll
    for (int j = 0; j < 8; ++j) {
      const int col = j * 16 + lp;
      float v = acc[j][i] + ws[WS_GW + g * 128 + col];
      v = fmaxf(v, 0.0f);                 // ReLU
      sc += v * W2[col];
    }
    sc = lane_reduce16(sc);
    if (lp == 0) out[r] = sc + b2v;
  }
}

extern "C" void kernel_launch(void* const* d_in, const int* in_sizes, int n_in,
                              void* d_out, int out_size, void* d_ws, size_t ws_size,
                              hipStream_t stream) {
  const float* feat  = (const float*)d_in[0];
  const int*   ei    = (const int*)d_in[1];
  const int*   batch = (const int*)d_in[2];
  const float* Wg    = (const float*)d_in[3];
  const float* att_s = (const float*)d_in[4];
  const float* att_d = (const float*)d_in[5];
  const float* bias  = (const float*)d_in[6];
  const float* W1    = (const float*)d_in[7];
  const float* b1    = (const float*)d_in[8];
  const float* W2    = (const float*)d_in[9];
  const float* b2    = (const float*)d_in[10];
  float* out = (float*)d_out;
  float* ws  = (float*)d_ws;

  k_init<<<(N_NODES * 128 + 255) / 256, 256, 0, stream>>>(ws);
  k_gat_gemm<<<N_NODES / 16, 256, 0, stream>>>(feat, Wg, att_s, att_d, ws);
  const int eb = (ETOT * 8 + 255) / 256;
  k_edge_max<<<eb, 256, 0, stream>>>(ei, ws);
  k_edge_sum<<<eb, 256, 0, stream>>>(ei, ws);
  k_edge_aggr<<<(ETOT * 32 + 255) / 256, 256, 0, stream>>>(ei, ws);
  k_finalize<<<(N_NODES + 31) / 32, 128, 0, stream>>>(batch, bias, ws);
  k_count<<<(N_NODES + 255) / 256, 256, 0, stream>>>(batch, ws);
  k_pool_div<<<(N_GRAPHS * 128 + 255) / 256, 256, 0, stream>>>(ws, out);
  k_gw<<<N_GRAPHS / 16, 256, 0, stream>>>(out + N_NODES, W1, b1, ws);
  k_decode<<<N_NODES / 16 / 5, 160, 0, stream>>>(W1, W2, b2, batch, ws, out);
}